// KroneckerLayer_31653908971736
// MI455X (gfx1250) — compile-verified
//
#include <hip/hip_runtime.h>

typedef __attribute__((ext_vector_type(2))) float v2f;
typedef __attribute__((ext_vector_type(8))) float v8f;

#define NN 8192
#define MM 8
#define DD 8
#define CC 32
#define KK 8
#define VROW (DD * CC)          // 256 floats per v row
#define OROW (MM * DD * CC)     // 2048 floats per output row
#define NCHUNKS 32
#define CHUNK (NN / NCHUNKS)    // 256 n per phase-1 block

// D = A(16x4) * B(4x16) + C, fp32 WMMA (CDNA5)
__device__ __forceinline__ v8f wmma4(v2f a, v2f b, v8f c) {
  return __builtin_amdgcn_wmma_f32_16x16x4_f32(
      /*neg_a=*/false, a, /*neg_b=*/false, b,
      /*c_mod=*/(short)0, c, /*reuse_a=*/false, /*reuse_b=*/false);
}

// ---------------------------------------------------------------------------
// Phase 1: S[m*8+d][c] = sum_n x[n,m] * v[n,d,c]   (raw sum; scaled in phase 2)
// GEMM view: S(8x256) = X^T(8xN) * V(NxDC), K=N; chunked WMMA + f32 atomics.
// Each wave owns one d (two 16-col tiles).
// ---------------------------------------------------------------------------
__global__ void __launch_bounds__(256) kron_sum_kernel(
    const float* __restrict__ x, const float* __restrict__ v,
    float* __restrict__ S) {
  const int lane = threadIdx.x & 31;
  const int lrow = lane & 15;
  const int half = lane >> 4;
  const int d    = threadIdx.x >> 5;        // wave id = d (0..7)
  const int nbase = blockIdx.x * CHUNK;
  const bool rowOk = (lrow < MM);

  // Per-lane base pointers; clamped row index keeps A loads unconditional.
  const float* xp0 = x + (long long)(nbase + 2 * half) * MM + (lrow & 7);
  const float* xp1 = xp0 + MM;                         // n0+1 (+2*half)
  const float* vp0 = v + (long long)(nbase + 2 * half) * VROW + d * CC + lrow;
  const float* vp1 = vp0 + VROW;

  v8f c0 = {};  // columns c = 0..15 of this d
  v8f c1 = {};  // columns c = 16..31 of this d
#pragma unroll 4
  for (int s = 0; s < CHUNK / 4; ++s) {
    v2f a, b0, b1;
    float t0 = xp0[0], t1 = xp1[0];        // always in-bounds (clamped)
    a.x = rowOk ? t0 : 0.0f;               // branchless v_cndmask
    a.y = rowOk ? t1 : 0.0f;
    b0.x = vp0[0];       b0.y = vp1[0];    // B[kk, c],  c = 0..15
    b1.x = vp0[16];      b1.y = vp1[16];   // B[kk, c], c = 16..31
    c0 = wmma4(a, b0, c0);
    c1 = wmma4(a, b1, c1);
    xp0 += 4 * MM;   xp1 += 4 * MM;
    vp0 += 4 * VROW; vp1 += 4 * VROW;
  }
  // Pack the two 16-col tiles into one 32-lane vector per row; atomic-add.
#pragma unroll
  for (int j = 0; j < MM; ++j) {
    float hi  = __shfl(c1[j], lane & 15, 32);
    float val = (lane < 16) ? c0[j] : hi;
    atomicAdd(&S[(j * DD + d) * CC + lane], val);
  }
}

// ---------------------------------------------------------------------------
// Phase 2: one wave per n.
// out[n,m,d,c] = th1[c]*x[n,m]*v[n,d,c] + th2[c]/N * S[md,c]
//             + th3[c]/K * sum_k x[nb_k,m]*v[nb_k,d,c]   (WMMA, K=8)
// ---------------------------------------------------------------------------
__global__ void __launch_bounds__(256) kron_main_kernel(
    const float* __restrict__ x, const float* __restrict__ v,
    const float* __restrict__ th1, const float* __restrict__ th2,
    const float* __restrict__ th3, const long long* __restrict__ nbr,
    const float* __restrict__ S, float* __restrict__ out) {
  const int lane = threadIdx.x & 31;
  const int lrow = lane & 15;
  const int half = lane >> 4;
  const int n    = blockIdx.x * 8 + (threadIdx.x >> 5);
  const bool rowOk = (lrow < MM);

  // channel c == lane (C==32): per-lane theta constants
  const float t1c = th1[lane];
  const float t2c = th2[lane] * (1.0f / (float)NN);
  const float t3c = th3[lane] * (1.0f / (float)KK);

  long long idx[KK];
#pragma unroll
  for (int k = 0; k < KK; ++k) idx[k] = nbr[(long long)n * KK + k];

  // A matrices for the two K=4 steps (shared across all d and both tiles).
  // Clamped row index -> unconditional loads, branchless zero of rows 8..15.
  v2f aA, aB;
  {
    const int cr = lrow & 7;
    float t0 = x[idx[0 + 2 * half] * MM + cr];
    float t1 = x[idx[1 + 2 * half] * MM + cr];
    float t2 = x[idx[4 + 2 * half] * MM + cr];
    float t3 = x[idx[5 + 2 * half] * MM + cr];
    aA.x = rowOk ? t0 : 0.0f;
    aA.y = rowOk ? t1 : 0.0f;
    aB.x = rowOk ? t2 : 0.0f;
    aB.y = rowOk ? t3 : 0.0f;
  }

  float xm[MM];
#pragma unroll
  for (int j = 0; j < MM; ++j) xm[j] = x[(long long)n * MM + j];

  // Per-lane base pointers (constant-offset addressing inside unrolled loop).
  const float* nb0 = v + idx[0 + 2 * half] * VROW + lrow;  // k-step0, vgpr.x
  const float* nb1 = v + idx[1 + 2 * half] * VROW + lrow;  // k-step0, vgpr.y
  const float* nb2 = v + idx[4 + 2 * half] * VROW + lrow;  // k-step1, vgpr.x
  const float* nb3 = v + idx[5 + 2 * half] * VROW + lrow;  // k-step1, vgpr.y
  const float* vnp = v + (long long)n * VROW + lane;
  const float* Sp  = S + lane;
  float*       onp = out + (long long)n * OROW + lane;

#pragma unroll
  for (int d = 0; d < DD; ++d) {
    v2f b0, b1;
    b0.x = nb0[d * CC];      b0.y = nb1[d * CC];
    b1.x = nb0[d * CC + 16]; b1.y = nb1[d * CC + 16];
    v8f c0 = {}, c1 = {};
    c0 = wmma4(aA, b0, c0);
    c1 = wmma4(aA, b1, c1);
    b0.x = nb2[d * CC];      b0.y = nb3[d * CC];
    b1.x = nb2[d * CC + 16]; b1.y = nb3[d * CC + 16];
    c0 = wmma4(aB, b0, c0);
    c1 = wmma4(aB, b1, c1);

    const float vs = vnp[d * CC];          // v[n,d,lane]
#pragma unroll
    for (int j = 0; j < MM; ++j) {
      float hi = __shfl(c1[j], lane & 15, 32);
      float g3 = (lane < 16) ? c0[j] : hi;     // neighbor-sum, row m=j
      float sv = Sp[(j * DD + d) * CC];        // global-sum tile (cache-hot)
      onp[(j * DD + d) * CC] =
          t1c * xm[j] * vs + t2c * sv + t3c * g3;  // 128B coalesced store
    }
  }
}

extern "C" void kernel_launch(void* const* d_in, const int* in_sizes, int n_in,
                              void* d_out, int out_size, void* d_ws, size_t ws_size,
                              hipStream_t stream) {
  const float* x        = (const float*)d_in[0];
  const float* v        = (const float*)d_in[1];
  const float* th1      = (const float*)d_in[2];
  const float* th2      = (const float*)d_in[3];
  const float* th3      = (const float*)d_in[4];
  const long long* nbr  = (const long long*)d_in[5];
  float* S   = (float*)d_ws;   // 64*32 f32 = 8KB scratch
  float* out = (float*)d_out;

  hipMemsetAsync(S, 0, MM * DD * CC * sizeof(float), stream);

  kron_sum_kernel<<<NCHUNKS, 256, 0, stream>>>(x, v, S);

  kron_main_kernel<<<NN / 8, 256, 0, stream>>>(x, v, th1, th2, th3, nbr, S, out);
}